// RogueWaveThreshold_25984552141475
// MI455X (gfx1250) — compile-verified
//
#include <hip/hip_runtime.h>
#include <math.h>

namespace {

constexpr int   BATCH = 32;
constexpr int   NELEM = 1024 * 1024;      // per-batch elements (2^20)
constexpr int   M4    = NELEM / 4;        // per-batch uint4/float4 count
constexpr unsigned KSEL = 349525u;        // max(1, N//3)
constexpr float STEEP = 10.0f;

constexpr int NB1 = 2048;                 // bits [31:21]
constexpr int NB2 = 2048;                 // bits [20:10]
constexpr int NB3 = 1024;                 // bits [9:0]
constexpr int CH  = 64;                   // histogram blocks per batch

// mask pipeline config
constexpr int MBLK = 2048;                // blocks
constexpr int MT   = 256;                 // threads per block

// workspace layout in 32-bit words
constexpr size_t OFF_H1    = 0;
constexpr size_t OFF_H2    = OFF_H1  + (size_t)BATCH * NB1;
constexpr size_t OFF_H3    = OFF_H2  + (size_t)BATCH * NB2;
constexpr size_t OFF_SEL   = OFF_H3  + (size_t)BATCH * NB3;   // 8 u32 per batch
constexpr size_t OFF_PARTH = OFF_SEL + (size_t)BATCH * 8;     // B*CH floats (S_hi partials)
constexpr size_t OFF_THR   = OFF_PARTH + (size_t)BATCH * CH;  // B floats

__device__ __forceinline__ float stable_sigmoid(float x) {
  float e = expf(-fabsf(x));
  float d = 1.0f + e;
  return (x >= 0.0f) ? (1.0f / d) : (e / d);
}

// ---- CDNA5 async global->LDS staging (ASYNCcnt-tracked, §15.18 / async_tensor)
__device__ __forceinline__ void async_load_b128(unsigned lds_addr,
                                                const void* gaddr) {
  asm volatile("global_load_async_to_lds_b128 %0, %1, off"
               :
               : "v"(lds_addr), "v"((unsigned long long)(size_t)gaddr)
               : "memory");
}
__device__ __forceinline__ void wait_async_le1() {
  asm volatile("s_wait_asynccnt 0x1" ::: "memory");
}
__device__ __forceinline__ void wait_async_le0() {
  asm volatile("s_wait_asynccnt 0x0" ::: "memory");
}

} // namespace

// ---------------------------------------------------------------- zero hist
__global__ void RW_zero_kernel(unsigned int* __restrict__ ws, int n) {
  int i = blockIdx.x * blockDim.x + threadIdx.x;
  if (i < n) ws[i] = 0u;
}

// ------------------------------------------------------- radix histograms
// LEVEL 3 additionally accumulates S_hi = sum of values whose 22-bit prefix
// is strictly greater than the cutoff prefix (deterministic block partials).
template <int LEVEL>
__global__ void RW_hist_kernel(const unsigned int* __restrict__ in,
                               unsigned int* ws) {
  constexpr int NB = (LEVEL == 3) ? NB3 : 2048;
  __shared__ unsigned int lh[NB];
  __shared__ float red[256];
  const int batch = blockIdx.y;
  for (int i = threadIdx.x; i < NB; i += blockDim.x) lh[i] = 0u;

  unsigned int p1 = 0u, p2 = 0u;
  if (LEVEL == 2) p1 = ws[OFF_SEL + (size_t)batch * 8 + 0];
  if (LEVEL == 3) p2 = ws[OFF_SEL + (size_t)batch * 8 + 2];
  __syncthreads();

  float shi = 0.0f;
  const uint4* pu = (const uint4*)(in + (size_t)batch * NELEM);
  const int stride = CH * blockDim.x;
  for (int j = blockIdx.x * blockDim.x + threadIdx.x; j < M4; j += stride) {
    __builtin_prefetch((const void*)(pu + j + stride), 0, 0);
    uint4 u = pu[j];
    if (LEVEL == 1) {
      atomicAdd(&lh[u.x >> 21], 1u);
      atomicAdd(&lh[u.y >> 21], 1u);
      atomicAdd(&lh[u.z >> 21], 1u);
      atomicAdd(&lh[u.w >> 21], 1u);
    } else if (LEVEL == 2) {
      if ((u.x >> 21) == p1) atomicAdd(&lh[(u.x >> 10) & 0x7FFu], 1u);
      if ((u.y >> 21) == p1) atomicAdd(&lh[(u.y >> 10) & 0x7FFu], 1u);
      if ((u.z >> 21) == p1) atomicAdd(&lh[(u.z >> 10) & 0x7FFu], 1u);
      if ((u.w >> 21) == p1) atomicAdd(&lh[(u.w >> 10) & 0x7FFu], 1u);
    } else {
      unsigned px = u.x >> 10, py = u.y >> 10, pz = u.z >> 10, pw = u.w >> 10;
      if (px == p2) atomicAdd(&lh[u.x & 0x3FFu], 1u);
      if (py == p2) atomicAdd(&lh[u.y & 0x3FFu], 1u);
      if (pz == p2) atomicAdd(&lh[u.z & 0x3FFu], 1u);
      if (pw == p2) atomicAdd(&lh[u.w & 0x3FFu], 1u);
      if (px > p2) shi += __uint_as_float(u.x);
      if (py > p2) shi += __uint_as_float(u.y);
      if (pz > p2) shi += __uint_as_float(u.z);
      if (pw > p2) shi += __uint_as_float(u.w);
    }
  }
  __syncthreads();

  unsigned int* gh =
      ws + (LEVEL == 1 ? OFF_H1 : (LEVEL == 2 ? OFF_H2 : OFF_H3)) +
      (size_t)batch * NB;
  for (int i = threadIdx.x; i < NB; i += blockDim.x)
    if (lh[i]) atomicAdd(&gh[i], lh[i]);

  if (LEVEL == 3) {
    // deterministic fixed-tree block reduction of S_hi
    red[threadIdx.x] = shi;
    __syncthreads();
    for (int s = 128; s > 0; s >>= 1) {
      if (threadIdx.x < s) red[threadIdx.x] += red[threadIdx.x + s];
      __syncthreads();
    }
    if (threadIdx.x == 0) {
      float* wsf = (float*)ws;
      wsf[OFF_PARTH + (size_t)batch * CH + blockIdx.x] = red[0];
    }
  }
}

// ------------------------------------------ radix scan for levels 1 and 2
__global__ void RW_scan_kernel(unsigned int* ws, int level) {
  const int b = threadIdx.x;
  if (b >= BATCH) return;
  unsigned int* sel = ws + OFF_SEL + (size_t)b * 8;

  const unsigned int* hist;
  int nb;
  unsigned int need;
  if (level == 1) { hist = ws + OFF_H1 + (size_t)b * NB1; nb = NB1; need = KSEL;   }
  else            { hist = ws + OFF_H2 + (size_t)b * NB2; nb = NB2; need = sel[1]; }

  unsigned int acc = 0u;
  int bin = nb - 1;
  int found = 0;
  for (; bin >= 0; --bin) {
    unsigned int c = hist[bin];
    if (acc + c >= need) { found = 1; break; }
    acc += c;
  }
  if (!found) bin = 0;

  if (level == 1) {
    sel[0] = (unsigned)bin;                  // 11-bit prefix
    sel[1] = need - acc;                     // still needed inside this bin
  } else {
    sel[2] = (sel[0] << 11) | (unsigned)bin; // 22-bit prefix
    sel[3] = need - acc;
  }
}

// ----- level-3 scan: finds v_k, folds in S_mid (exact from counts, since a
// level-3 bin is a single bit pattern) and S_hi partials -> final threshold
__global__ void RW_scan3_kernel(unsigned int* ws, float* __restrict__ out) {
  const int b = threadIdx.x;
  if (b >= BATCH) return;
  unsigned int* sel = ws + OFF_SEL + (size_t)b * 8;
  const unsigned int* hist = ws + OFF_H3 + (size_t)b * NB3;
  const unsigned int need = sel[3];
  const unsigned int p2   = sel[2];

  unsigned int acc = 0u;
  float smid = 0.0f;
  int bin = NB3 - 1;
  int found = 0;
  for (; bin >= 0; --bin) {
    unsigned int c = hist[bin];
    if (acc + c >= need) { found = 1; break; }
    if (c) {
      acc += c;
      smid += (float)c * __uint_as_float((p2 << 10) | (unsigned)bin);
    }
  }
  if (!found) bin = 0;

  const float vk = __uint_as_float((p2 << 10) | (unsigned)bin);
  const unsigned int take = need - acc;      // copies of v_k still needed

  // deterministic fixed-order reduction of the CH block partials
  const float* part = (const float*)ws + OFF_PARTH + (size_t)b * CH;
  float shi = 0.0f;
  for (int i = 0; i < CH; ++i) shi += part[i];

  const float total = shi + smid + (float)take * vk;   // exact top-k sum
  const float thr   = 2.0f * (total / (float)KSEL);
  out[(size_t)BATCH * NELEM + b] = thr;                // threshold output
  ((float*)ws)[OFF_THR + b] = thr;                     // for mask pass
}

// ------------- mask pass: async-LDS double-buffered streaming pipeline
__global__ void RW_mask_kernel(const float4* __restrict__ in4,
                               float* __restrict__ out,
                               const unsigned int* __restrict__ ws) {
  __shared__ float4 stage[2][MT];
  const float* wsf = (const float*)ws;

  const size_t total4   = (size_t)BATCH * NELEM / 4;
  const size_t perBlock = total4 / MBLK;               // 4096 float4
  const size_t base     = (size_t)blockIdx.x * perBlock;
  const int    tiles    = (int)(perBlock / MT);        // 16

  // each block stays inside one batch (16384 elems divides 2^20)
  const int   batch = (int)((base * 4) >> 20);
  const float thr   = wsf[OFF_THR + batch];

  const unsigned lds0 = (unsigned)(size_t)&stage[0][threadIdx.x];
  const unsigned lds1 = (unsigned)(size_t)&stage[1][threadIdx.x];

  float4* o0 = (float4*)out;
  float4* o2 = (float4*)(out + (size_t)BATCH * NELEM + BATCH);

  // prime the pipeline
  async_load_b128(lds0, in4 + base + threadIdx.x);

  for (int t = 0; t < tiles; ++t) {
    if (t + 1 < tiles) {
      async_load_b128(((t + 1) & 1) ? lds1 : lds0,
                      in4 + base + (size_t)(t + 1) * MT + threadIdx.x);
      wait_async_le1();   // oldest (current tile) has landed in LDS
    } else {
      wait_async_le0();
    }

    float4 v = stage[t & 1][threadIdx.x];
    float4 m;
    m.x = stable_sigmoid(STEEP * (v.x - thr));
    m.y = stable_sigmoid(STEEP * (v.y - thr));
    m.z = stable_sigmoid(STEEP * (v.z - thr));
    m.w = stable_sigmoid(STEEP * (v.w - thr));

    const size_t i = base + (size_t)t * MT + threadIdx.x;
    o0[i] = m;
    o2[i] = m;
  }
}

// ---------------------------------------------------------------- launcher
extern "C" void kernel_launch(void* const* d_in, const int* in_sizes, int n_in,
                              void* d_out, int out_size, void* d_ws,
                              size_t ws_size, hipStream_t stream) {
  (void)in_sizes; (void)n_in; (void)out_size; (void)ws_size;

  const float*        in  = (const float*)d_in[0];
  const unsigned int* inu = (const unsigned int*)d_in[0];
  float*        out = (float*)d_out;
  unsigned int* ws  = (unsigned int*)d_ws;

  // 1) zero the three histogram regions (everything else is overwritten)
  {
    const int nzero = (int)OFF_SEL;
    RW_zero_kernel<<<(nzero + 255) / 256, 256, 0, stream>>>(ws, nzero);
  }

  // 2-6) three-level radix select; level 3 also produces the top-k sum
  RW_hist_kernel<1><<<dim3(CH, BATCH), 256, 0, stream>>>(inu, ws);
  RW_scan_kernel<<<1, 32, 0, stream>>>(ws, 1);
  RW_hist_kernel<2><<<dim3(CH, BATCH), 256, 0, stream>>>(inu, ws);
  RW_scan_kernel<<<1, 32, 0, stream>>>(ws, 2);
  RW_hist_kernel<3><<<dim3(CH, BATCH), 256, 0, stream>>>(inu, ws);
  RW_scan3_kernel<<<1, 32, 0, stream>>>(ws, out);

  // 7) sigmoid mask via async-LDS pipeline, written to both output slots
  RW_mask_kernel<<<MBLK, MT, 0, stream>>>((const float4*)in, out, ws);
}